// BiAttention_7318624272773
// MI455X (gfx1250) — compile-verified
//
#include <hip/hip_runtime.h>
#include <hip/hip_bf16.h>
#include <cstddef>

typedef float v2f __attribute__((ext_vector_type(2)));
typedef float v8f __attribute__((ext_vector_type(8)));

#define VERY_NEG (-1e30f)

union U8 { v8f v; float f[8]; };

// One workgroup per (b,m): 256 blocks x 512 threads (16 waves of 32).
// LDS: query (raw + w3-weighted) 2x33.8KB, per-wave 16x68 logit/prob tiles
// 69.6KB, plus reductions => ~147KB (<160KB => 2 WGs per 320KB WGP).
__global__ __launch_bounds__(512)
void biattn_fused_kernel(const float* __restrict__ text,        // (256, 512, 128)
                         const float* __restrict__ query,       // (32, 64, 128)
                         const float* __restrict__ text_mask,   // (256, 512)
                         const float* __restrict__ query_mask,  // (32, 64)
                         const float* __restrict__ w,           // (384,)
                         const float* __restrict__ bias,        // (1,)
                         float* __restrict__ out)               // (256, 512, 512)
{
    const int JX = 512, D = 128, C = 512;

    __shared__ float qraw[64 * 132];     // query[j][d], padded
    __shared__ float qwl [64 * 132];     // query[j][d]*w3[d] + w1[d]
    __shared__ float ltile[16 * 16 * 68];// per-wave 16x64 logits -> probs
    __shared__ float qlmax_s[512];
    __shared__ float pt_s[512];
    __shared__ float red_s[512];
    __shared__ float tap_s[512];
    __shared__ float tm_s[512];
    __shared__ float qm_s[64];
    __shared__ float q2_s[64];
    __shared__ float ta_s[128];

    const int tid = threadIdx.x;
    const int bm  = blockIdx.x;          // 0..255  (b*8 + m)
    const int bb  = bm >> 3;
    const float bval = bias[0];

    const float* textBM = text  + (size_t)bm * JX * D;
    const float* qB     = query + (size_t)bb * 64 * D;
    const size_t obase  = (size_t)bm * JX * C;

    // ---- stage query (raw and w-fused), masks ----
    for (int r = 0; r < 16; ++r) {
        int idx = r * 512 + tid;           // 0..8191 = j*128+d
        int j = idx >> 7, d = idx & 127;
        float qv = qB[idx];
        qraw[j * 132 + d] = qv;
        qwl [j * 132 + d] = qv * w[256 + d] + w[d];   // w3*q + w1  (folds t1)
    }
    tm_s[tid] = text_mask[(size_t)bm * JX + tid];
    if (tid < 64) qm_s[tid] = query_mask[bb * 64 + tid];
    __syncthreads();

    // ---- q2[j] = query[j] . w2 ----
    if (tid < 64) {
        float s = 0.f;
        for (int d = 0; d < 128; ++d) s += qraw[tid * 132 + d] * w[128 + d];
        q2_s[tid] = s;
    }
    __syncthreads();

    const int lane = tid & 31;
    const int wv   = tid >> 5;           // wave 0..15
    const int ln   = lane & 15;
    const int kb0  = (lane >> 4) * 2;    // f32 WMMA A/B lane k-base
    const bool hi  = lane >= 16;
    float* myt = &ltile[wv * 1088];      // 16 x 68 per wave

    for (int half = 0; half < 2; ++half) {
        const int rowbase = wv * 32 + half * 16;

        // ---- GEMM1: (cross + t1)[16x64] = text[16x128] @ qwl^T ----
        U8 c[4];
        for (int jt = 0; jt < 4; ++jt)
            for (int v = 0; v < 8; ++v) c[jt].f[v] = 0.f;

        const float* trow = textBM + (size_t)(rowbase + ln) * D;
        for (int k = 0; k < 128; k += 4) {
            int kb = k + kb0;
            v2f a  = *(const v2f*)(trow + kb);
            v2f b0 = *(const v2f*)&qwl[( 0 + ln) * 132 + kb];
            v2f b1 = *(const v2f*)&qwl[(16 + ln) * 132 + kb];
            v2f b2 = *(const v2f*)&qwl[(32 + ln) * 132 + kb];
            v2f b3 = *(const v2f*)&qwl[(48 + ln) * 132 + kb];
            c[0].v = __builtin_amdgcn_wmma_f32_16x16x4_f32(false, a, false, b0, (short)0, c[0].v, false, false);
            c[1].v = __builtin_amdgcn_wmma_f32_16x16x4_f32(false, a, false, b1, (short)0, c[1].v, false, false);
            c[2].v = __builtin_amdgcn_wmma_f32_16x16x4_f32(false, a, false, b2, (short)0, c[2].v, false, false);
            c[3].v = __builtin_amdgcn_wmma_f32_16x16x4_f32(false, a, false, b3, (short)0, c[3].v, false, false);
        }

        // ---- epilogue: + q2 + bias + mask, write logits tile to LDS ----
        for (int jt = 0; jt < 4; ++jt) {
            int col = jt * 16 + ln;
            float colterm = q2_s[col] + bval;
            float qmv = qm_s[col];
            for (int v = 0; v < 8; ++v) {
                int r = hi ? v + 8 : v;
                int i = rowbase + r;
                myt[r * 68 + col] =
                    c[jt].f[v] + colterm + (1.0f - tm_s[i] * qmv) * VERY_NEG;
            }
        }

        // ---- per-row softmax over j (64) + qlmax ----
        if (!hi) {
            float* lrow = &myt[ln * 68];
            float mx = -3.4e38f;
            for (int j = 0; j < 64; ++j) mx = fmaxf(mx, lrow[j]);
            qlmax_s[rowbase + ln] = mx;
            float s = 0.f;
            for (int j = 0; j < 64; ++j) { float e = __expf(lrow[j] - mx); lrow[j] = e; s += e; }
            float inv = 1.f / s;
            for (int j = 0; j < 64; ++j) lrow[j] *= inv;
        }
        __syncthreads();   // uniform: both halves, all waves

        // ---- GEMM2: query_attn[16x128] = P[16x64] @ qraw[64x128] ----
        U8 q[8];
        for (int nt = 0; nt < 8; ++nt)
            for (int v = 0; v < 8; ++v) q[nt].f[v] = 0.f;

        for (int k = 0; k < 64; k += 4) {
            int kb = k + kb0;
            v2f a = *(const v2f*)&myt[ln * 68 + kb];
            for (int nt = 0; nt < 8; ++nt) {
                int n = nt * 16 + ln;
                v2f bfr;
                bfr.x = qraw[ kb      * 132 + n];
                bfr.y = qraw[(kb + 1) * 132 + n];
                q[nt].v = __builtin_amdgcn_wmma_f32_16x16x4_f32(false, a, false, bfr, (short)0, q[nt].v, false, false);
            }
        }

        // ---- write out segments 0..2 ----
        for (int nt = 0; nt < 8; ++nt) {
            int cc = nt * 16 + ln;
            for (int v = 0; v < 8; ++v) {
                int r = hi ? v + 8 : v;
                int i = rowbase + r;
                float qa = q[nt].f[v];
                float tv = textBM[(size_t)i * D + cc];
                float* orow = out + obase + (size_t)i * C;
                orow[cc]        = tv;          // text
                orow[128 + cc]  = qa;          // query_attn
                orow[256 + cc]  = tv * qa;     // text * query_attn
            }
        }
    }
    __syncthreads();

    // ---- softmax over i of qlmax -> p_text ----
    red_s[tid] = qlmax_s[tid];
    __syncthreads();
    for (int s = 256; s > 0; s >>= 1) {
        if (tid < s) red_s[tid] = fmaxf(red_s[tid], red_s[tid + s]);
        __syncthreads();
    }
    float gmax = red_s[0];
    __syncthreads();
    float e = __expf(qlmax_s[tid] - gmax);
    red_s[tid] = e;
    __syncthreads();
    for (int s = 256; s > 0; s >>= 1) {
        if (tid < s) red_s[tid] += red_s[tid + s];
        __syncthreads();
    }
    float inv = 1.f / red_s[0];
    pt_s[tid] = e * inv;
    __syncthreads();

    // ---- text_attn[d] = sum_i p_text[i] * text[i][d] ----
    {
        int d = tid & 127, g = tid >> 7;
        const float* tcol = textBM + (size_t)g * 128 * D + d;
        const float* ptp  = &pt_s[g * 128];
        float acc = 0.f;
        for (int r = 0; r < 128; ++r) acc += ptp[r] * tcol[(size_t)r * D];
        tap_s[g * 128 + d] = acc;
    }
    __syncthreads();
    if (tid < 128)
        ta_s[tid] = tap_s[tid] + tap_s[128 + tid] + tap_s[256 + tid] + tap_s[384 + tid];
    __syncthreads();

    // ---- segment 3: text * text_attn (coalesced stream) ----
    for (int r = 0; r < 128; ++r) {
        int idx = r * 512 + tid;         // i*128 + d
        int i = idx >> 7, d = idx & 127;
        out[obase + (size_t)i * C + 384 + d] = textBM[idx] * ta_s[d];
    }
}

extern "C" void kernel_launch(void* const* d_in, const int* in_sizes, int n_in,
                              void* d_out, int out_size, void* d_ws, size_t ws_size,
                              hipStream_t stream) {
    (void)in_sizes; (void)n_in; (void)out_size; (void)d_ws; (void)ws_size;
    const float* text       = (const float*)d_in[0];
    const float* query      = (const float*)d_in[1];
    const float* text_mask  = (const float*)d_in[2];
    const float* query_mask = (const float*)d_in[3];
    const float* w          = (const float*)d_in[4];
    const float* b          = (const float*)d_in[5];
    float* out = (float*)d_out;
    biattn_fused_kernel<<<256, 512, 0, stream>>>(text, query, text_mask, query_mask, w, b, out);
}